// OP_83468394430932
// MI455X (gfx1250) — compile-verified
//
#include <hip/hip_runtime.h>
#include <hip/hip_bf16.h>
#include <stdint.h>

// ---------------- problem constants (from reference) ----------------
#define FPS_B       64
#define FPS_N       131072
#define FPS_NPOINT  512
#define FPS_T       1024                      // threads per block (32 waves, wave32)
#define FPS_TILE    2048                      // points per chunk/tile (2 per thread)
#define FPS_NCHUNK  27                        // chunks with dist pinned in LDS
#define FPS_DPIN    (FPS_NCHUNK * FPS_TILE)   // 55296 points -> 216 KB LDS
#define FPS_REST    (FPS_N - FPS_DPIN)        // 75776 points, dist in global ws
#define FPS_NTILE   (FPS_REST / FPS_TILE)     // 37 streamed xyz tiles
#define FPS_NBUF    4                         // async LDS tile ring (96 KB)

// ---------------- CDNA5 async-to-LDS path (probe-confirmed available) -------
#if defined(__AMDGCN__) && __has_builtin(__builtin_amdgcn_global_load_async_to_lds_b64)
#define FPS_ASYNC 1
#else
#define FPS_ASYNC 0
#endif

#if FPS_ASYNC
// builtin signature (from clang diagnostic): param0 = v2i in AS(1) ("__device__"),
// param1 = v2i in AS(3), then (imm) offset, (imm) cpol.
typedef int fps_v2i __attribute__((__vector_size__(8)));
typedef __attribute__((address_space(1))) fps_v2i fps_gv2i;
typedef __attribute__((address_space(3))) fps_v2i fps_lv2i;

#if __has_builtin(__builtin_amdgcn_s_wait_asynccnt)
#define FPS_WAIT_ASYNC(n)                                   \
  do {                                                      \
    __builtin_amdgcn_s_wait_asynccnt(n);                    \
    asm volatile("" ::: "memory");                          \
  } while (0)
#else
#define FPS_WAIT_ASYNC(n) asm volatile("s_wait_asynccnt %0" ::"i"(n) : "memory")
#endif

// one b64 async copy: 8 bytes from global to this lane's own LDS slot
#define FPS_ASYNC_B64(g, l)                                 \
  __builtin_amdgcn_global_load_async_to_lds_b64(            \
      (fps_gv2i*)(g), (fps_lv2i*)(l), 0, 0)
#endif

// (max value, min index) combine -> matches jnp.argmax first-occurrence tie-break
__device__ __forceinline__ void fps_combine(float& v, int& i, float ov, int oi) {
  if (ov > v || (ov == v && oi < i)) { v = ov; i = oi; }
}

__device__ __forceinline__ void fps_wave_reduce(float& v, int& i) {
#pragma unroll
  for (int m = 16; m > 0; m >>= 1) {
    float ov = __shfl_xor(v, m, 32);
    int   oi = __shfl_xor(i, m, 32);
    fps_combine(v, i, ov, oi);
  }
}

__global__ __launch_bounds__(FPS_T) void fps_gather_kernel(
    const float* __restrict__ seeds,  // (B, 3, N)
    float* __restrict__ out,          // (B, 3, NPOINT)
    float* __restrict__ distg)        // (B, FPS_REST) scratch
{
  const int b = blockIdx.x;
  const int t = threadIdx.x;
  const float* xs = seeds + (size_t)b * 3u * FPS_N;
  const float* ys = xs + FPS_N;
  const float* zs = xs + 2 * FPS_N;
  float* dist2 = distg + (size_t)b * FPS_REST;
  float* outb  = out + (size_t)b * 3u * FPS_NPOINT;

  // 216 KB: running min-sq-dist for points [0, FPS_DPIN) pinned in LDS
  __shared__ float s_dist[FPS_DPIN];
#if FPS_ASYNC
  // 96 KB: 4-deep ring of xyz tiles for the streamed region
  __shared__ float s_tile[FPS_NBUF][3][FPS_TILE];
#endif
  __shared__ float s_last[3];
  __shared__ float s_wv[FPS_T / 32];
  __shared__ int   s_wi[FPS_T / 32];
  __shared__ int   s_next;

  const int lane = t & 31;
  const int wave = t >> 5;
  const int o    = 2 * t;  // this thread's pair offset inside a 2048-pt chunk

#if FPS_ASYNC
  // producer == consumer: each lane async-copies exactly the pairs it will
  // read, so per-wave s_wait_asynccnt suffices (no per-tile block barrier).
#define FPS_ISSUE_TILE(jj)                                          \
  do {                                                              \
    const int _j = (jj);                                            \
    const int _bb = _j & (FPS_NBUF - 1);                            \
    const int _n = FPS_DPIN + _j * FPS_TILE + o;                    \
    FPS_ASYNC_B64(xs + _n, &s_tile[_bb][0][o]);                     \
    FPS_ASYNC_B64(ys + _n, &s_tile[_bb][1][o]);                     \
    FPS_ASYNC_B64(zs + _n, &s_tile[_bb][2][o]);                     \
  } while (0)
#endif

  int last = 0;  // reference starts at index 0

  for (int i = 0; i < FPS_NPOINT; ++i) {
    // broadcast coords of the current pick through LDS
    if (t == 0) {
      s_last[0] = xs[last];
      s_last[1] = ys[last];
      s_last[2] = zs[last];
    }
    __syncthreads();
    const float lx = s_last[0], ly = s_last[1], lz = s_last[2];
    // gather output: column i is the PRE-update pick (scan semantics)
    if (t < 3) outb[t * FPS_NPOINT + i] = s_last[t];

#if FPS_ASYNC
    // prime the async pipeline: tiles 0..3 in flight (12 b64 per lane)
#pragma unroll
    for (int j = 0; j < FPS_NBUF; ++j) FPS_ISSUE_TILE(j);
#endif

    float bestv = -1.0f;
    int   besti = 0;

    // ---- region 1: dist pinned in LDS, xyz streamed from global (L2) ----
#pragma unroll 3
    for (int k = 0; k < FPS_NCHUNK; ++k) {
      const int n = k * FPS_TILE + o;
      const float2 px = *(const float2*)(xs + n);
      const float2 py = *(const float2*)(ys + n);
      const float2 pz = *(const float2*)(zs + n);
      const float dx0 = px.x - lx, dy0 = py.x - ly, dz0 = pz.x - lz;
      const float dx1 = px.y - lx, dy1 = py.y - ly, dz1 = pz.y - lz;
      const float d0 = dx0 * dx0 + dy0 * dy0 + dz0 * dz0;
      const float d1 = dx1 * dx1 + dy1 * dy1 + dz1 * dz1;
      const float o0 = (i == 0) ? 1e10f : s_dist[n];
      const float o1 = (i == 0) ? 1e10f : s_dist[n + 1];
      const float nd0 = fminf(o0, d0);
      const float nd1 = fminf(o1, d1);
      s_dist[n]     = nd0;
      s_dist[n + 1] = nd1;
      if (nd0 > bestv) { bestv = nd0; besti = n; }
      if (nd1 > bestv) { bestv = nd1; besti = n + 1; }
    }

    // ---- region 2: xyz via async LDS ring, dist in global scratch ----
#pragma unroll 1
    for (int j = 0; j < FPS_NTILE; ++j) {
      const int r = j * FPS_TILE + o;  // index in streamed region
      const int n = FPS_DPIN + r;      // global point index
      float x0, y0, z0, x1, y1, z1;
#if FPS_ASYNC
      // in-order completion: leave 3 outstanding per still-in-flight future tile
      if (j + 3 < FPS_NTILE)      FPS_WAIT_ASYNC(9);
      else if (j + 2 < FPS_NTILE) FPS_WAIT_ASYNC(6);
      else if (j + 1 < FPS_NTILE) FPS_WAIT_ASYNC(3);
      else                        FPS_WAIT_ASYNC(0);
      const int bb = j & (FPS_NBUF - 1);
      x0 = s_tile[bb][0][o]; x1 = s_tile[bb][0][o + 1];
      y0 = s_tile[bb][1][o]; y1 = s_tile[bb][1][o + 1];
      z0 = s_tile[bb][2][o]; z1 = s_tile[bb][2][o + 1];
#else
      const float2 px = *(const float2*)(xs + n);
      const float2 py = *(const float2*)(ys + n);
      const float2 pz = *(const float2*)(zs + n);
      x0 = px.x; x1 = px.y; y0 = py.x; y1 = py.y; z0 = pz.x; z1 = pz.y;
#endif
      const float dx0 = x0 - lx, dy0 = y0 - ly, dz0 = z0 - lz;
      const float dx1 = x1 - lx, dy1 = y1 - ly, dz1 = z1 - lz;
      const float d0 = dx0 * dx0 + dy0 * dy0 + dz0 * dz0;
      const float d1 = dx1 * dx1 + dy1 * dy1 + dz1 * dz1;
      float o0, o1;
      if (i == 0) {
        o0 = 1e10f; o1 = 1e10f;  // never read poisoned scratch on first pass
      } else {
        const float2 od = *(const float2*)(dist2 + r);
        o0 = od.x; o1 = od.y;
      }
      const float nd0 = fminf(o0, d0);
      const float nd1 = fminf(o1, d1);
      *(float2*)(dist2 + r) = make_float2(nd0, nd1);
      if (nd0 > bestv) { bestv = nd0; besti = n; }
      if (nd1 > bestv) { bestv = nd1; besti = n + 1; }
#if FPS_ASYNC
      if (j + FPS_NBUF < FPS_NTILE) FPS_ISSUE_TILE(j + FPS_NBUF);  // refill freed buf
#endif
    }

    // ---- block argmax: wave32 butterfly, then 32 wave leaders ----
    fps_wave_reduce(bestv, besti);
    if (lane == 0) { s_wv[wave] = bestv; s_wi[wave] = besti; }
    __syncthreads();
    if (wave == 0) {
      bestv = s_wv[lane];
      besti = s_wi[lane];
      fps_wave_reduce(bestv, besti);
      if (lane == 0) s_next = besti;
    }
    __syncthreads();
    last = s_next;
  }
}

extern "C" void kernel_launch(void* const* d_in, const int* in_sizes, int n_in,
                              void* d_out, int out_size, void* d_ws, size_t ws_size,
                              hipStream_t stream) {
  const float* seeds = (const float*)d_in[0];  // (64, 3, 131072)
  // d_in[1] = sg_input: dead code in the reference branch, intentionally unused
  (void)in_sizes; (void)n_in; (void)out_size; (void)ws_size;
  float* out = (float*)d_out;                  // (64, 3, 512)
  float* ws  = (float*)d_ws;                   // 64 * 75776 floats (~19.4 MB)
  fps_gather_kernel<<<dim3(FPS_B), dim3(FPS_T), 0, stream>>>(seeds, out, ws);
}